// TripletLoss_80195629351538
// MI455X (gfx1250) — compile-verified
//
#include <hip/hip_runtime.h>
#include <hip/hip_bf16.h>

// Problem constants (from reference): B=128, V=4, D=128 -> N=512, 32 classes.
#define BATCH 128
#define VIEWS 4
#define DIM   128
#define NTOT  (BATCH * VIEWS)   // 512

typedef float v2f __attribute__((ext_vector_type(2)));
typedef float v8f __attribute__((ext_vector_type(8)));

// ---------------------------------------------------------------------------
// Kernel 1: restack (view-major), L2-normalize rows, emit per-row sq-norm.
// One wave32 per embedding row. Also zeroes the global accumulators.
// ---------------------------------------------------------------------------
__global__ void tl_normalize(const float* __restrict__ features,
                             float* __restrict__ embN,   // [N, D]
                             float* __restrict__ sn,     // [N]
                             float* __restrict__ acc)    // [2] {sum, cnt}
{
    if (blockIdx.x == 0 && threadIdx.x == 0) { acc[0] = 0.0f; acc[1] = 0.0f; }

    const int wave = threadIdx.x >> 5;          // 8 waves per block
    const int lane = threadIdx.x & 31;
    const int row  = blockIdx.x * 8 + wave;     // 64 blocks * 8 = 512 rows
    if (row >= NTOT) return;

    // emb[row] = features[b, v, :] with b = row % 128, v = row / 128
    const int b = row & (BATCH - 1);
    const int v = row >> 7;
    const float* src = features + ((size_t)b * VIEWS + v) * DIM;

    float x0 = src[lane];
    float x1 = src[lane + 32];
    float x2 = src[lane + 64];
    float x3 = src[lane + 96];

    float ss = x0 * x0 + x1 * x1 + x2 * x2 + x3 * x3;
    #pragma unroll
    for (int off = 16; off > 0; off >>= 1)
        ss += __shfl_xor(ss, off, 32);

    const float inv = 1.0f / fmaxf(sqrtf(ss), 1e-12f);

    float* dst = embN + (size_t)row * DIM;
    dst[lane]      = x0 * inv;
    dst[lane + 32] = x1 * inv;
    dst[lane + 64] = x2 * inv;
    dst[lane + 96] = x3 * inv;

    if (lane == 0) sn[row] = ss * inv * inv;    // == sum(embN[row]^2)
}

// ---------------------------------------------------------------------------
// Kernel 2: pairwise distance matrix via f32 WMMA (V_WMMA_F32_16X16X4_F32).
// One wave32 per 16x16 output tile; K-loop of 32 steps over D=128.
// A fragment (16x4 f32): lanes 0-15 hold M=0..15 with K={0,1} in v{0,1};
// lanes 16-31 hold the same M with K={2,3}.  B (4x16) mirrors with N on lanes.
// ---------------------------------------------------------------------------
__global__ void tl_gram_dist(const float* __restrict__ embN,
                             const float* __restrict__ sn,
                             float* __restrict__ Dm)     // [N, N]
{
    const int lane = threadIdx.x & 31;
    const int i0 = blockIdx.y * 16;             // tile row base
    const int j0 = blockIdx.x * 16;             // tile col base

    const int m  = lane & 15;                   // M (or N) index for this lane
    const int kh = (lane >> 4) * 2;             // K sub-offset for this half

    const float* arow = embN + (size_t)(i0 + m) * DIM + kh;
    const float* brow = embN + (size_t)(j0 + m) * DIM + kh;

    v8f acc = {};
    #pragma unroll 4
    for (int k = 0; k < DIM; k += 4) {
        v2f a = *(const v2f*)(arow + k);
        v2f b = *(const v2f*)(brow + k);
        acc = __builtin_amdgcn_wmma_f32_16x16x4_f32(
            /*neg_a=*/false, a, /*neg_b=*/false, b,
            /*c_mod=*/(short)0, acc, /*reuse_a=*/false, /*reuse_b=*/false);
    }

    // C/D layout: VGPR r, lanes 0-15 -> (M=r, N=lane); lanes 16-31 -> (M=r+8).
    const int jj = j0 + (lane & 15);
    const int mb = i0 + ((lane >> 4) * 8);
    const float snj = sn[jj];
    #pragma unroll
    for (int r = 0; r < 8; ++r) {
        const int ii = mb + r;
        const float sq = sn[ii] + snj - 2.0f * acc[r];
        Dm[(size_t)ii * NTOT + jj] = (sq > 0.0f) ? sqrtf(sq) : 0.0f;
    }
}

// ---------------------------------------------------------------------------
// Kernel 3: triplet reduction. One block (256 threads) per anchor.
// loss[a,p,n] = relu(d_ap - d_an) over {lab_p==lab_a, p!=a, lab_n!=lab_a}.
// Accumulate sum and nonzero-count (margin=0 -> nonzero iff d_ap > d_an).
// ---------------------------------------------------------------------------
__global__ void tl_triplet(const float* __restrict__ Dm,
                           const int* __restrict__ labels,  // [128]
                           float* __restrict__ acc)         // [2]
{
    __shared__ float sD[NTOT];
    __shared__ int   sLab[BATCH];
    __shared__ float rs[256];
    __shared__ float rc[256];

    const int a = blockIdx.x;
    const int t = threadIdx.x;

    sD[t]       = Dm[(size_t)a * NTOT + t];
    sD[t + 256] = Dm[(size_t)a * NTOT + t + 256];
    if (t < BATCH) sLab[t] = labels[t];
    __syncthreads();

    const int aLab = sLab[a & (BATCH - 1)];

    // This thread's two negatives candidates: n0=t, n1=t+256 (same label slot).
    const bool isNeg = (sLab[t & (BATCH - 1)] != aLab);
    const float dn0 = sD[t];
    const float dn1 = sD[t + 256];

    float s = 0.0f, c = 0.0f;
    for (int p = 0; p < NTOT; ++p) {
        // Block-uniform mask: positives share anchor's label, p != a.
        if (p == a || sLab[p & (BATCH - 1)] != aLab) continue;
        const float dap = sD[p];
        if (isNeg) {
            if (dap > dn0) { s += dap - dn0; c += 1.0f; }
            if (dap > dn1) { s += dap - dn1; c += 1.0f; }
        }
    }

    rs[t] = s; rc[t] = c;
    __syncthreads();
    #pragma unroll
    for (int off = 128; off > 0; off >>= 1) {
        if (t < off) { rs[t] += rs[t + off]; rc[t] += rc[t + off]; }
        __syncthreads();
    }
    if (t == 0) {
        atomicAdd(&acc[0], rs[0]);
        atomicAdd(&acc[1], rc[0]);
    }
}

// ---------------------------------------------------------------------------
// Kernel 4: finalize (AvgNonZeroReducer).
// ---------------------------------------------------------------------------
__global__ void tl_finalize(const float* __restrict__ acc, float* __restrict__ out)
{
    if (threadIdx.x == 0)
        out[0] = (acc[1] > 0.0f) ? (acc[0] / acc[1]) : 0.0f;
}

// ---------------------------------------------------------------------------
extern "C" void kernel_launch(void* const* d_in, const int* in_sizes, int n_in,
                              void* d_out, int out_size, void* d_ws, size_t ws_size,
                              hipStream_t stream)
{
    (void)in_sizes; (void)n_in; (void)out_size; (void)ws_size;

    const float* features = (const float*)d_in[0];   // [128,4,128] f32
    const int*   labels   = (const int*)d_in[1];     // [128] i32

    float* ws   = (float*)d_ws;
    float* embN = ws;                                 // 512*128 = 65536 f
    float* sn   = embN + (size_t)NTOT * DIM;          // 512 f
    float* Dm   = sn + NTOT;                          // 512*512 = 262144 f
    float* acc  = Dm + (size_t)NTOT * NTOT;           // 2 f  (~1.28 MB total)

    tl_normalize<<<64, 256, 0, stream>>>(features, embN, sn, acc);
    tl_gram_dist<<<dim3(NTOT / 16, NTOT / 16), 32, 0, stream>>>(embN, sn, Dm);
    tl_triplet<<<NTOT, 256, 0, stream>>>(Dm, labels, acc);
    tl_finalize<<<1, 1, 0, stream>>>(acc, (float*)d_out);
}